// Int16SiLULUT_30983894073633
// MI455X (gfx1250) — compile-verified
//
#include <hip/hip_runtime.h>
#include <stdint.h>

// Int16 SiLU via Q8.8 LUT — memory-bound streaming kernel for gfx1250.
// CDNA5 paths used: global_load_async_to_lds_b128 + s_wait_asynccnt (LUT stage),
// b128 vmem with non-temporal hints, ds_load_u16 gathers from LDS.

typedef float f32x4 __attribute__((ext_vector_type(4)));
typedef int   v4i   __attribute__((ext_vector_type(4)));
typedef __attribute__((address_space(1))) v4i gv4i;  // global (AS1) int4
typedef __attribute__((address_space(3))) v4i lv4i;  // LDS (AS3) int4

#define TPB 256

__device__ __forceinline__ float silu_q88(float xv, const uint16_t* __restrict__ lut) {
    // Quantize: round-half-even, clamp to int16 range (matches jnp.round+clip).
    float t = rintf(xv * 256.0f);                  // v_rndne_f32
    t = fminf(fmaxf(t, -32768.0f), 32767.0f);
    int xq = (int)t;                               // exact: integral & in range
    // idx = clamp(xq, -2048, 2048) + 2048  (exact collapse of the float index math)
    int idx = xq < -2048 ? -2048 : (xq > 2048 ? 2048 : xq);
    int s = (int)lut[idx + 2048];                  // ds_load_u16; s in [0, 256]
    int prod = xq * s;                             // |prod| <= 2^23, fits int32
    // Round-to-nearest-even >> 8 (branch-free; equals sign-magnitude RNE)
    int y = (prod + 127 + ((prod >> 8) & 1)) >> 8; // fits int16 by construction
    return (float)y * (1.0f / 256.0f);
}

__global__ __launch_bounds__(TPB) void silu_lut_kernel(
    const float* __restrict__ x, const int16_t* __restrict__ table,
    float* __restrict__ out, int n4, int n) {
    __shared__ uint16_t lut[4104];                 // 4097 entries, padded

    const int t = threadIdx.x;
    // --- Stage LUT[0..4095] (8192 B) into LDS with CDNA5 async copy ---
    {
#if __has_builtin(__builtin_amdgcn_global_load_async_to_lds_b128)
        gv4i* g = (gv4i*)((const char*)table + t * 32);   // AS1, non-const per builtin sig
        lv4i* l = (lv4i*)((char*)&lut[0]   + t * 32);     // AS3
        __builtin_amdgcn_global_load_async_to_lds_b128(g, l, 0, 0);
        __builtin_amdgcn_global_load_async_to_lds_b128(g, l, 16, 0); // offset applies to both sides
#else
        uint32_t laddr = (uint32_t)(uintptr_t)(const void*)&lut[0] + (uint32_t)t * 32u;
        const char* gaddr = (const char*)table + t * 32;
        asm volatile(
            "global_load_async_to_lds_b128 %0, %1, off\n\t"
            "global_load_async_to_lds_b128 %0, %1, off offset:16"
            :: "v"(laddr), "v"(gaddr) : "memory");
#endif
#if __has_builtin(__builtin_amdgcn_s_wait_asynccnt)
        __builtin_amdgcn_s_wait_asynccnt(0);
#else
        asm volatile("s_wait_asynccnt 0" ::: "memory");
#endif
    }
    if (t == 0) lut[4096] = (uint16_t)table[4096]; // last entry via plain load
    __syncthreads();

    // --- Streaming main loop: float4 per thread, grid-stride, non-temporal ---
    const int i0 = blockIdx.x * TPB + t;
    const int stride = gridDim.x * TPB;
    const f32x4* __restrict__ x4 = (const f32x4*)x;
    f32x4* __restrict__ o4 = (f32x4*)out;
    for (int i = i0; i < n4; i += stride) {
        f32x4 v = __builtin_nontemporal_load(&x4[i]);
        f32x4 r;
        r.x = silu_q88(v.x, lut);
        r.y = silu_q88(v.y, lut);
        r.z = silu_q88(v.z, lut);
        r.w = silu_q88(v.w, lut);
        __builtin_nontemporal_store(r, &o4[i]);
    }
    // Tail (n not divisible by 4; not hit for this problem size but kept correct)
    for (int i = n4 * 4 + i0; i < n; i += stride) {
        out[i] = silu_q88(x[i], lut);
    }
}

extern "C" void kernel_launch(void* const* d_in, const int* in_sizes, int n_in,
                              void* d_out, int out_size, void* d_ws, size_t ws_size,
                              hipStream_t stream) {
    const float*   x     = (const float*)d_in[0];
    const int16_t* table = (const int16_t*)d_in[1];
    float*         out   = (float*)d_out;
    int n  = in_sizes[0];
    int n4 = n / 4;

    int blocks = (n4 + TPB - 1) / TPB;
    if (blocks > 4096) blocks = 4096;   // grid-stride; ~16 WGs/WGP-class occupancy
    if (blocks < 1)    blocks = 1;

    silu_lut_kernel<<<blocks, TPB, 0, stream>>>(x, table, out, n4, n);
}